// MorseModel_27238682592043
// MI455X (gfx1250) — compile-verified
//
#include <hip/hip_runtime.h>
#include <stdint.h>

// ---------------- problem constants (match reference) ----------------
#define SIGMA_C   1.0f
#define EPS_C     5.0f
#define ALPHA_C   5.0f
#define CUTOFF_C  2.5f
#define N_ATOMS_C 262144
#define N_PAIRS_C 8388608

// ---------------- launch geometry ----------------
#define TPB      256                        // 8 wave32 waves per block
#define PPT      4                          // pairs per thread per tile
#define TILE     (TPB * PPT)                // 1024 pairs per tile
#define NBLOCKS  2048                       // 8192 tiles total
#define TILES_PER_BLOCK (N_PAIRS_C / TILE / NBLOCKS)   // exactly 4

// ---------------- LDS layout (40 KB -> 8 blocks/WGP on 320 KB) ----------------
struct alignas(16) SMem {
    int   ib[2][TILE];        // 8 KB   mapping[0] (i indices)
    int   jb[2][TILE];        // 8 KB   mapping[1] (j indices)
    float sb[2][TILE * 3];    // 24 KB  shifts
    // reduction scratch (10*TPB floats = 10 KB) aliases ib/jb after the
    // pipeline drains (guarded by __syncthreads()).
};

// Flat->LDS address: on AMDGCN the low 32 bits of a generic pointer into LDS
// are the DS byte address (addrspacecast flat->local is a truncation).
__device__ __forceinline__ uint32_t lds_off(const void* p) {
    return (uint32_t)(uintptr_t)p;
}

// CDNA5 async copy: global -> LDS, 16B/lane, ASYNCcnt-tracked.  TH_LOAD_NT:
// the pair stream is read exactly once (168 MB), so keep it from evicting the
// hot ~7 MB gather/atomic working set out of the 192 MB L2.
__device__ __forceinline__ void async_b128(uint32_t ldsoff, const void* gaddr) {
    asm volatile("global_load_async_to_lds_b128 %0, %1, off th:TH_LOAD_NT"
                 :: "v"(ldsoff), "v"(gaddr) : "memory");
}

template <int N>
__device__ __forceinline__ void wait_asynccnt() {
    asm volatile("s_wait_asynccnt %0" :: "n"(N) : "memory");
}

__device__ __forceinline__ void atomic_add_f32(float* p, float v) {
    __hip_atomic_fetch_add(p, v, __ATOMIC_RELAXED, __HIP_MEMORY_SCOPE_AGENT);
}

// Issue the 5 async b128 loads that fetch THIS thread's PPT pairs of a tile.
// Each thread stages only data it will itself consume -> a per-wave
// s_wait_asynccnt is sufficient; no workgroup barrier in the pipeline.
__device__ __forceinline__ void issue_tile(const int* __restrict__ mapping,
                                           const float* __restrict__ shifts,
                                           int tile, int buf, int tid, SMem* sm) {
    const int p0 = tile * TILE + tid * PPT;                 // 4 contiguous pairs
    async_b128(lds_off(&sm->ib[buf][tid * PPT]), mapping + p0);
    async_b128(lds_off(&sm->jb[buf][tid * PPT]), mapping + N_PAIRS_C + p0);
    const float* sp = shifts + (size_t)p0 * 3;              // 48B, 16B aligned
    const uint32_t so = lds_off(&sm->sb[buf][tid * PPT * 3]);
    async_b128(so,      sp);
    async_b128(so + 16, sp + 4);
    async_b128(so + 32, sp + 8);
}

__global__ void __launch_bounds__(TPB)
morse_pairs(const float* __restrict__ positions,   // [N_ATOMS,3]
            const float* __restrict__ cell,        // [3,3]
            const float* __restrict__ shifts,      // [N_PAIRS,3]
            const int*   __restrict__ mapping,     // [2,N_PAIRS]
            float* __restrict__ out_energy,        // [1]
            float* __restrict__ atom_e,            // [N_ATOMS]
            float* __restrict__ forces,            // [N_ATOMS,3]
            float* __restrict__ out_stress)        // [3,3]
{
    __shared__ SMem sm;
    const int tid = threadIdx.x;

    // Uniform cell -> scalar loads, lives in SGPRs.
    const float c00 = cell[0], c01 = cell[1], c02 = cell[2];
    const float c10 = cell[3], c11 = cell[4], c12 = cell[5];
    const float c20 = cell[6], c21 = cell[7], c22 = cell[8];

    float esum = 0.0f;
    float s00 = 0.f, s01 = 0.f, s02 = 0.f;
    float s10 = 0.f, s11 = 0.f, s12 = 0.f;
    float s20 = 0.f, s21 = 0.f, s22 = 0.f;

    // One Morse pair: all inputs in registers, accumulators captured.
    auto body = [&](int ai, int aj, float shx, float shy, float shz) {
        const float* pi = positions + 3 * ai;      // global_load_b96 (L2 hit)
        const float* pj = positions + 3 * aj;
        const float pix = pi[0], piy = pi[1], piz = pi[2];
        const float pjx = pj[0], pjy = pj[1], pjz = pj[2];

        // shifts @ cell (row-vector times 3x3)
        const float dsx = shx * c00 + shy * c10 + shz * c20;
        const float dsy = shx * c01 + shy * c11 + shz * c21;
        const float dsz = shx * c02 + shy * c12 + shz * c22;

        const float dx = pjx - pix + dsx;
        const float dy = pjy - piy + dsy;
        const float dz = pjz - piz + dsz;

        const float r2 = dx * dx + dy * dy + dz * dz;
        const float r  = sqrtf(r2);
        const bool mask = (r < CUTOFF_C) && (r > 1e-10f);

        const float e  = expf(-ALPHA_C * (r - SIGMA_C));
        const float om = 1.0f - e;
        const float pe = mask ? (EPS_C * om * om - EPS_C) : 0.0f;
        const float pf = mask ? (-2.0f * ALPHA_C * EPS_C * e * om) : 0.0f;
        const float safe_r = (r > 1e-10f) ? r : 1.0f;
        const float fs = pf / safe_r;
        const float fx = fs * dx, fy = fs * dy, fz = fs * dz;

        esum += pe;
        s00 += dx * fx; s01 += dx * fy; s02 += dx * fz;
        s10 += dy * fx; s11 += dy * fy; s12 += dy * fz;
        s20 += dz * fx; s21 += dz * fy; s22 += dz * fz;

        if (mask) {   // skip dead atomics for the ~half of pairs beyond cutoff
            const float he = 0.5f * pe;
            atomic_add_f32(atom_e + ai, he);
            atomic_add_f32(atom_e + aj, he);
            atomic_add_f32(forces + 3 * ai + 0, -fx);
            atomic_add_f32(forces + 3 * ai + 1, -fy);
            atomic_add_f32(forces + 3 * ai + 2, -fz);
            atomic_add_f32(forces + 3 * aj + 0,  fx);
            atomic_add_f32(forces + 3 * aj + 1,  fy);
            atomic_add_f32(forces + 3 * aj + 2,  fz);
        }
    };

    // Software pipeline, fully unrolled (exactly TILES_PER_BLOCK iterations,
    // gridDim.x == NBLOCKS): compile-time buffer indices & LDS immediates.
    issue_tile(mapping, shifts, blockIdx.x, 0, tid, &sm);

#pragma unroll
    for (int k = 0; k < TILES_PER_BLOCK; ++k) {
        const int t   = blockIdx.x + k * NBLOCKS;
        const int buf = k & 1;
        if (k + 1 < TILES_PER_BLOCK) {
            issue_tile(mapping, shifts, t + NBLOCKS, buf ^ 1, tid, &sm);
            wait_asynccnt<5>();   // 5 newer outstanding -> current tile landed
        } else {
            wait_asynccnt<0>();
        }

        // This thread's staged tile data: 5x ds_load_b128, all registers.
        const int4   iv = *(const int4*)  &sm.ib[buf][tid * PPT];
        const int4   jv = *(const int4*)  &sm.jb[buf][tid * PPT];
        const float4 sA = *(const float4*)&sm.sb[buf][tid * PPT * 3];
        const float4 sB = *(const float4*)&sm.sb[buf][tid * PPT * 3 + 4];
        const float4 sC = *(const float4*)&sm.sb[buf][tid * PPT * 3 + 8];

        body(iv.x, jv.x, sA.x, sA.y, sA.z);
        body(iv.y, jv.y, sA.w, sB.x, sB.y);
        body(iv.z, jv.z, sB.z, sB.w, sC.x);
        body(iv.w, jv.w, sC.y, sC.z, sC.w);
    }

    // ---- block reduction of energy + stress (reuse staging LDS) ----
    __syncthreads();                       // all staging reads complete
    float* red = (float*)&sm;              // 10*TPB floats alias ib/jb
    red[0 * TPB + tid] = esum;
    red[1 * TPB + tid] = s00; red[2 * TPB + tid] = s01; red[3 * TPB + tid] = s02;
    red[4 * TPB + tid] = s10; red[5 * TPB + tid] = s11; red[6 * TPB + tid] = s12;
    red[7 * TPB + tid] = s20; red[8 * TPB + tid] = s21; red[9 * TPB + tid] = s22;
    __syncthreads();

    if (tid < 10) {
        float acc = 0.0f;
        for (int k = 0; k < TPB; ++k) acc += red[tid * TPB + k];
        if (tid == 0) {
            atomic_add_f32(out_energy, 0.5f * acc);   // energy = 0.5 * sum(pe)
        } else {
            const float det = c00 * (c11 * c22 - c12 * c21)
                            - c01 * (c10 * c22 - c12 * c20)
                            + c02 * (c10 * c21 - c11 * c20);
            const float vol = fabsf(det);
            atomic_add_f32(out_stress + (tid - 1), -acc / vol);
        }
    }
}

extern "C" void kernel_launch(void* const* d_in, const int* in_sizes, int n_in,
                              void* d_out, int out_size, void* d_ws, size_t ws_size,
                              hipStream_t stream) {
    const float* positions = (const float*)d_in[0];   // [262144,3]  f32
    const float* cell      = (const float*)d_in[1];   // [3,3]       f32
    const float* shifts    = (const float*)d_in[2];   // [8388608,3] f32
    const int*   mapping   = (const int*)d_in[3];     // [2,8388608] i32
    float* out = (float*)d_out;

    // d_out layout: [0] energy | [1 .. N_ATOMS] atom_energies |
    //               [1+N_ATOMS .. 1+4*N_ATOMS) forces | last 9 = stress
    float* out_energy = out;
    float* atom_e     = out + 1;
    float* forces     = out + 1 + N_ATOMS_C;
    float* out_stress = out + 1 + N_ATOMS_C + 3 * N_ATOMS_C;

    hipMemsetAsync(d_out, 0, (size_t)out_size * sizeof(float), stream);

    morse_pairs<<<NBLOCKS, TPB, 0, stream>>>(positions, cell, shifts, mapping,
                                             out_energy, atom_e, forces, out_stress);
}